// EqualizeClahe_63299228009134
// MI455X (gfx1250) — compile-verified
//
#include <hip/hip_runtime.h>

// ---------------------------------------------------------------------------
// CLAHE for x:(2,3,2048,2048) f32, grid 8x8, 256 bins, clip 40.
// Kernel 1: one block per tile -> LDS wave-private histograms (ds_add_u32),
//           clip+redistribute, then a 256-element inclusive scan done with
//           8 chained V_WMMA_F32_16X16X4_F32 (exact in f32), LUT -> d_ws.
// Kernel 2: streaming bilinear LUT apply (float4 loads/stores, LUT gathers
//           hit L2; the whole input is L2-resident after kernel 1).
// ---------------------------------------------------------------------------

typedef __attribute__((ext_vector_type(2))) float v2f;
typedef __attribute__((ext_vector_type(8))) float v8f;

#define NBINS   256
#define WIDTH   2048
#define HEIGHT  2048
#define NCH     6            // B*C = 2*3
#define TILEW   256
#define TILEH   256
#define PIXELS  65536.0f     // per tile
#define CLIPMAX 10240.0f     // max(int(40*65536)//256, 1)

__global__ __launch_bounds__(256)
void clahe_hist_lut(const float* __restrict__ x, float* __restrict__ lut) {
    __shared__ unsigned int hist[8][NBINS];   // one copy per wave32
    __shared__ float histF[NBINS];
    __shared__ float Tlds[16];                // per-row totals for scan
    __shared__ float sumC;

    const int tid  = threadIdx.x;
    const int wave = tid >> 5;                // wave32
    const int tile = blockIdx.x;              // ((bc*8+gy)*8+gx) == reference order
    const int tx = tile & 7;
    const int ty = (tile >> 3) & 7;
    const int bc = tile >> 6;

    for (int i = tid; i < 8 * NBINS; i += 256) ((unsigned int*)hist)[i] = 0u;
    if (tid == 0) sumC = 0.0f;
    __syncthreads();

    // ---- phase 1: wave-private LDS histograms (coalesced float4 stream) ----
    const float* base = x + (size_t)bc * ((size_t)HEIGHT * WIDTH)
                          + (size_t)(ty * TILEH) * WIDTH + tx * TILEW;
    const int c4 = (tid & 63) << 2;           // column within tile (float4)
    const int r0 = tid >> 6;                  // 0..3
    for (int it = 0; it < 64; ++it) {
        const int row = r0 + (it << 2);
        const float4 v = *(const float4*)(base + (size_t)row * WIDTH + c4);
        int b0 = min(255, max(0, (int)floorf(v.x * 256.0f)));
        int b1 = min(255, max(0, (int)floorf(v.y * 256.0f)));
        int b2 = min(255, max(0, (int)floorf(v.z * 256.0f)));
        int b3 = min(255, max(0, (int)floorf(v.w * 256.0f)));
        atomicAdd(&hist[wave][b0], 1u);
        atomicAdd(&hist[wave][b1], 1u);
        atomicAdd(&hist[wave][b2], 1u);
        atomicAdd(&hist[wave][b3], 1u);
    }
    __syncthreads();

    // ---- phase 2: merge + clip + redistribute ----
    {
        unsigned int cnt = 0;
        #pragma unroll
        for (int w = 0; w < 8; ++w) cnt += hist[w][tid];
        const float c = fminf((float)cnt, CLIPMAX);
        histF[tid] = c;
        atomicAdd(&sumC, c);                  // ds_add_f32; exact (ints <= 2^16)
    }
    __syncthreads();
    {
        const float excess = (PIXELS - sumC) * (1.0f / 256.0f);
        histF[tid] += excess;                 // multiples of 1/256, exact in f32
    }
    __syncthreads();

    // ---- phase 3: 256-wide inclusive scan via WMMA f32 16x16x4 (wave 0) ----
    // View histF as H[16][16] row-major. P = H * U (U upper-tri ones) gives
    // intra-row inclusive scans; S = Lstrict * rowTotals + P adds row offsets.
    if (wave == 0) {
        const int l  = tid;                   // 0..31
        const int m  = l & 15;                // A row / B,C,D column index
        const int hi = l >> 4;                // half-wave select

        v8f P = {};
        #pragma unroll
        for (int k = 0; k < 4; ++k) {
            const int kc = 4 * k + 2 * hi;    // K-pair handled by this half-wave
            v2f a, b;
            a.x = histF[m * 16 + kc];         // A = H[:, kc:kc+2]
            a.y = histF[m * 16 + kc + 1];
            b.x = (kc     <= m) ? 1.0f : 0.0f; // B = U[kc:kc+2, n=m]
            b.y = (kc + 1 <= m) ? 1.0f : 0.0f;
            P = __builtin_amdgcn_wmma_f32_16x16x4_f32(
                    false, a, false, b, (short)0, P, false, false);
        }

        // row totals t[r] = P[r][15] live in lanes 15/31; share via LDS
        if (m == 15) {
            #pragma unroll
            for (int g = 0; g < 8; ++g) Tlds[g + 8 * hi] = P[g];
        }
        asm volatile("s_wait_dscnt 0" ::: "memory");

        v8f S = P;
        #pragma unroll
        for (int k = 0; k < 4; ++k) {
            const int kc = 4 * k + 2 * hi;
            v2f a, b;
            a.x = (kc     < m) ? 1.0f : 0.0f; // A = Lstrict[:, kc:kc+2]
            a.y = (kc + 1 < m) ? 1.0f : 0.0f;
            b.x = Tlds[kc];                   // B row kc broadcast = t[kc]
            b.y = Tlds[kc + 1];
            S = __builtin_amdgcn_wmma_f32_16x16x4_f32(
                    false, a, false, b, (short)0, S, false, false);
        }

        // LUT = floor(clip(cumsum * 255/65536, 0, 255)); exact f32 throughout
        const float lutScale = 255.0f / 65536.0f;
        float* lt = lut + (size_t)tile * NBINS;
        #pragma unroll
        for (int g = 0; g < 8; ++g) {
            const int r = g + 8 * hi;         // C/D layout: VGPR g -> rows g, g+8
            const float s = fminf(fmaxf(S[g] * lutScale, 0.0f), 255.0f);
            lt[r * 16 + m] = floorf(s);
        }
    }
}

__global__ __launch_bounds__(256)
void clahe_apply(const float* __restrict__ x, const float* __restrict__ lut,
                 float* __restrict__ out) {
    const size_t i4   = (size_t)blockIdx.x * 256 + threadIdx.x;
    const size_t flat = i4 << 2;
    const int col = (int)(flat & (WIDTH - 1));
    const int row = (int)((flat >> 11) & (HEIGHT - 1));
    const int bc  = (int)(flat >> 22);

    const float4 v = *(const float4*)(x + flat);

    const float ys = fminf(fmaxf(((float)row + 0.5f) * (1.0f / 256.0f) - 0.5f,
                                 0.0f), 7.0f);
    const int   y0 = (int)ys;                 // ys >= 0 -> trunc == floor
    const float wy = ys - (float)y0;
    const int   y1 = min(y0 + 1, 7);

    const float* lb = lut + (size_t)bc * (64 * NBINS);
    const float* l0 = lb + y0 * 8 * NBINS;
    const float* l1 = lb + y1 * 8 * NBINS;

    const float vv[4] = {v.x, v.y, v.z, v.w};
    float res[4];
    #pragma unroll
    for (int j = 0; j < 4; ++j) {
        const float xs = fminf(fmaxf(((float)(col + j) + 0.5f) * (1.0f / 256.0f)
                                     - 0.5f, 0.0f), 7.0f);
        const int   x0 = (int)xs;
        const float wx = xs - (float)x0;
        const int   x1 = min(x0 + 1, 7);
        const int   p  = min(255, max(0, (int)floorf(vv[j] * 255.0f)));
        const float v00 = l0[x0 * NBINS + p];
        const float v01 = l0[x1 * NBINS + p];
        const float v10 = l1[x0 * NBINS + p];
        const float v11 = l1[x1 * NBINS + p];
        res[j] = ((1.0f - wy) * ((1.0f - wx) * v00 + wx * v01) +
                  wy         * ((1.0f - wx) * v10 + wx * v11)) / 255.0f;
    }
    *(float4*)(out + flat) = make_float4(res[0], res[1], res[2], res[3]);
}

extern "C" void kernel_launch(void* const* d_in, const int* in_sizes, int n_in,
                              void* d_out, int out_size, void* d_ws, size_t ws_size,
                              hipStream_t stream) {
    const float* x   = (const float*)d_in[0];
    float*       out = (float*)d_out;
    float*       lut = (float*)d_ws;          // 384 * 256 floats = 384 KB

    clahe_hist_lut<<<NCH * 64, 256, 0, stream>>>(x, lut);

    const int n4     = (NCH * HEIGHT * WIDTH) / 4;   // 6291456 float4s
    const int blocks = n4 / 256;                      // 24576
    clahe_apply<<<blocks, 256, 0, stream>>>(x, lut, out);
}